// ChamferLoss_46918222742368
// MI455X (gfx1250) — compile-verified
//
#include <hip/hip_runtime.h>
#include <math.h>

// Problem shape from the reference: B=4, D=3, N=8192.
#define NB 4
#define NP 8192
#define WG_THREADS 256                         // 8 waves of 32
#define ROWS_PER_WAVE 16
#define WAVES_PER_WG (WG_THREADS / 32)
#define ROWS_PER_WG (ROWS_PER_WAVE * WAVES_PER_WG)   // 128
#define JSPLIT 4                               // split gt sweep across 4 WGs
#define JCOLS (NP / JSPLIT)                    // 2048 gt columns per WG
#define JTILES (JCOLS / 16)                    // 128 tiles per wave
#define PF 8                                   // software-pipeline depth (tiles)

typedef __attribute__((ext_vector_type(2))) float v2f;
typedef __attribute__((ext_vector_type(8))) float v8f;

// ---------------------------------------------------------------------------
// Kernel 1: precompute ||p||^2, ||g||^2; init global colmin/rowmin to +inf
// ---------------------------------------------------------------------------
__global__ void chamfer_init(const float* __restrict__ pred,
                             const float* __restrict__ gt,
                             float* __restrict__ psq,
                             float* __restrict__ gsq,
                             unsigned int* __restrict__ colmin,
                             unsigned int* __restrict__ rowmin) {
    int idx = blockIdx.x * blockDim.x + threadIdx.x;   // 0 .. NB*NP-1
    if (idx >= NB * NP) return;
    int b = idx / NP;
    int n = idx % NP;
    const float* pb = pred + (size_t)b * 3 * NP;
    const float* gb = gt   + (size_t)b * 3 * NP;
    float px = pb[n], py = pb[NP + n], pz = pb[2 * NP + n];
    psq[idx] = px * px + py * py + pz * pz;
    float gx = gb[n], gy = gb[NP + n], gz = gb[2 * NP + n];
    gsq[idx] = gx * gx + gy * gy + gz * gz;
    colmin[idx] = 0x7F800000u;   // +inf bits (valid uint ordering for >=0 floats)
    rowmin[idx] = 0x7F800000u;
}

// ---------------------------------------------------------------------------
// Main sweep. WG = (batch b, 128 pred rows, 1/4 of the gt columns).
// Each wave owns 16 pred rows; one V_WMMA_F32_16X16X4_F32 per 16x16 tile:
//   d2[m][n] = [-2p,1]·[g,||g||^2] + ||p||^2
// B operands stream through a 2-deep ping-pong of 8-tile register blocks so
// global loads stay ~200+ cycles ahead of their consuming WMMA.
// ---------------------------------------------------------------------------
__global__ void __launch_bounds__(WG_THREADS)
chamfer_main(const float* __restrict__ pred,
             const float* __restrict__ gt,
             const float* __restrict__ psq,
             const float* __restrict__ gsq,
             unsigned int* __restrict__ colmin,   // [NB*NP] float-bits
             unsigned int* __restrict__ rowmin) { // [NB*NP] float-bits
    __shared__ unsigned int lds_col[JCOLS];       // 8 KB staging for col-mins

    const int tid = threadIdx.x;
    for (int k = tid; k < JCOLS; k += WG_THREADS) lds_col[k] = 0x7F800000u;
    __syncthreads();

    const int wave = tid >> 5;
    const int lane = tid & 31;
    const int half = lane >> 4;     // 0: lanes 0-15, 1: lanes 16-31
    const int col  = lane & 15;

    int wg = blockIdx.x;                         // NB * (NP/128) * JSPLIT
    const int jblk = wg % JSPLIT;  wg /= JSPLIT;
    const int iblk = wg % (NP / ROWS_PER_WG);
    const int b    = wg / (NP / ROWS_PER_WG);
    const int i0   = iblk * ROWS_PER_WG + wave * ROWS_PER_WAVE;
    const int jbase = jblk * JCOLS;

    const float* pb  = pred + (size_t)b * 3 * NP;
    const float* gb  = gt   + (size_t)b * 3 * NP;
    const float* psb = psq  + (size_t)b * NP;
    const float* gsb = gsq  + (size_t)b * NP;

    // ---- A (16x4): row m = [-2px, -2py, -2pz, 1]; M = lane%16,
    //      vgpr0: K0 (lo lanes)/K2 (hi lanes); vgpr1: K1/K3.
    const int mA = i0 + col;
    v2f a;
    a.x = -2.0f * pb[(half ? 2 : 0) * NP + mA];
    a.y = half ? 1.0f : (-2.0f * pb[NP + mA]);

    // ---- C init: ||p||^2 broadcast along N; C/D: M = r + 8*half.
    v8f cini;
#pragma unroll
    for (int r = 0; r < 8; ++r) cini[r] = psb[i0 + r + half * 8];

    // ---- per-lane B stream pointers (loop-invariant; stride 16 floats/tile)
    const float* bx_ptr = gb + (half ? 2 * NP : 0) + jbase + col;    // gx | gz
    const float* by_ptr = (half ? gsb : (gb + NP)) + jbase + col;    // gy | ||g||^2

    float rowacc[8];
#pragma unroll
    for (int r = 0; r < 8; ++r) rowacc[r] = INFINITY;

    v2f bufA[PF], bufB[PF];
#pragma unroll
    for (int k = 0; k < PF; ++k) { bufA[k].x = bx_ptr[k * 16]; bufA[k].y = by_ptr[k * 16]; }

    // process one 8-tile block held in registers
    auto process = [&](const v2f* buf, int t0) {
#pragma unroll
        for (int k = 0; k < PF; k += 2) {
            v8f dA = __builtin_amdgcn_wmma_f32_16x16x4_f32(
                false, a, false, buf[k],     (short)0, cini, false, false);
            v8f dB = __builtin_amdgcn_wmma_f32_16x16x4_f32(
                false, a, false, buf[k + 1], (short)0, cini, false, false);
#pragma unroll
            for (int r = 0; r < 8; ++r)               // -> v_min3_num_f32
                rowacc[r] = fminf(rowacc[r], fminf(dA[r], dB[r]));
            // column mins of each tile (min over 16 M-rows, both halves merge
            // in the LDS atomic since lanes l and l^16 share an address)
            float cA = fminf(fminf(fminf(dA[0], dA[1]), fminf(dA[2], dA[3])),
                             fminf(fminf(dA[4], dA[5]), fminf(dA[6], dA[7])));
            float cB = fminf(fminf(fminf(dB[0], dB[1]), fminf(dB[2], dB[3])),
                             fminf(fminf(dB[4], dB[5]), fminf(dB[6], dB[7])));
            cA = fmaxf(cA, 0.0f);
            cB = fmaxf(cB, 0.0f);
            atomicMin(&lds_col[(t0 + k) * 16 + col],     __float_as_uint(cA));
            atomicMin(&lds_col[(t0 + k + 1) * 16 + col], __float_as_uint(cB));
        }
    };

    for (int t0 = 0; t0 < JTILES; t0 += 2 * PF) {
        // prefetch block t0+PF while computing block t0
#pragma unroll
        for (int k = 0; k < PF; ++k) {
            int t = t0 + PF + k;
            bufB[k].x = bx_ptr[t * 16]; bufB[k].y = by_ptr[t * 16];
        }
        process(bufA, t0);
        if (t0 + 2 * PF < JTILES) {     // uniform branch
#pragma unroll
            for (int k = 0; k < PF; ++k) {
                int t = t0 + 2 * PF + k;
                bufA[k].x = bx_ptr[t * 16]; bufA[k].y = by_ptr[t * 16];
            }
        }
        process(bufB, t0 + PF);
    }

    // ---- row-mins: reduce across the 16 lanes of each half, then global min
#pragma unroll
    for (int r = 0; r < 8; ++r) {
        float v = rowacc[r];
        v = fminf(v, __shfl_xor(v, 1, 32));
        v = fminf(v, __shfl_xor(v, 2, 32));
        v = fminf(v, __shfl_xor(v, 4, 32));
        v = fminf(v, __shfl_xor(v, 8, 32));
        if (col == 0)
            atomicMin(&rowmin[(size_t)b * NP + i0 + r + half * 8],
                      __float_as_uint(fmaxf(v, 0.0f)));
    }

    __syncthreads();
    // ---- drain staged col-mins to global (once per WG)
    for (int k = tid; k < JCOLS; k += WG_THREADS)
        atomicMin(&colmin[(size_t)b * NP + jbase + k], lds_col[k]);
}

// ---------------------------------------------------------------------------
// Kernel 3: loss = (sum sqrt(colmin) + sum sqrt(rowmin)) / B
// ---------------------------------------------------------------------------
__global__ void chamfer_finalize(const unsigned int* __restrict__ colmin,
                                 const unsigned int* __restrict__ rowmin,
                                 float* __restrict__ out) {
    __shared__ float red[256];
    const int tid = threadIdx.x;
    float s = 0.0f;
    for (int k = tid; k < NB * NP; k += 256) {
        s += sqrtf(__uint_as_float(colmin[k]));   // z : min over pred, per gt
        s += sqrtf(__uint_as_float(rowmin[k]));   // z2: min over gt, per pred
    }
    red[tid] = s;
    __syncthreads();
    for (int off = 128; off > 0; off >>= 1) {
        if (tid < off) red[tid] += red[tid + off];
        __syncthreads();
    }
    if (tid == 0) out[0] = red[0] / (float)NB;
}

// ---------------------------------------------------------------------------
extern "C" void kernel_launch(void* const* d_in, const int* in_sizes, int n_in,
                              void* d_out, int out_size, void* d_ws, size_t ws_size,
                              hipStream_t stream) {
    (void)in_sizes; (void)n_in; (void)out_size; (void)ws_size;
    const float* pred = (const float*)d_in[0];
    const float* gt   = (const float*)d_in[1];
    float* out = (float*)d_out;

    // workspace: psq | gsq | colmin(uint) | rowmin(uint)  = 512 KB
    float* psq           = (float*)d_ws;
    float* gsq           = psq + NB * NP;
    unsigned int* colmin = (unsigned int*)(gsq + NB * NP);
    unsigned int* rowmin = colmin + NB * NP;

    chamfer_init<<<(NB * NP + 255) / 256, 256, 0, stream>>>(pred, gt, psq, gsq,
                                                            colmin, rowmin);
    chamfer_main<<<NB * (NP / ROWS_PER_WG) * JSPLIT, WG_THREADS, 0, stream>>>(
        pred, gt, psq, gsq, colmin, rowmin);
    chamfer_finalize<<<1, 256, 0, stream>>>(colmin, rowmin, out);
}